// MS2DGBlock_80272938762543
// MI455X (gfx1250) — compile-verified
//
#include <hip/hip_runtime.h>

// Problem constants (match reference)
#define BB    4
#define CC    128
#define NN    8192
#define KNN   20
#define OUTC  128
#define TWOC  256

typedef __attribute__((ext_vector_type(2))) float v2f;
typedef __attribute__((ext_vector_type(8))) float v8f;

#define NEG_INF (-3.402823466e38f)
#define POS_INF ( 3.402823466e38f)

// ---------------------------------------------------------------------------
// Kernel 0: xx[b,n] = sum_c x[b,c,n]^2
// ---------------------------------------------------------------------------
__global__ void __launch_bounds__(256) norms_kernel(const float* __restrict__ x,
                                                    float* __restrict__ xx) {
    int i = blockIdx.x * 256 + threadIdx.x;   // over B*N
    if (i >= BB * NN) return;
    int b = i / NN, n = i % NN;
    const float* xp = x + (size_t)b * CC * NN + n;
    float s = 0.f;
#pragma unroll 8
    for (int c = 0; c < CC; ++c) {
        float v = xp[(size_t)c * NN];
        s += v * v;
    }
    xx[i] = s;
}

// ---------------------------------------------------------------------------
// Kernel 1: Y1[b,o,n] = (W1+W2)x + bias   (layout [B][OUT][N])
//           Y2t[b,n,o] = W2 x             (layout [B][N][OUT] for gathers)
// One wave computes a 16(o) x 16(n) tile with WMMA f32 16x16x4, K=C=128.
// ---------------------------------------------------------------------------
__global__ void __launch_bounds__(32) y_gemm_kernel(const float* __restrict__ x,
                                                    const float* __restrict__ W,
                                                    const float* __restrict__ bias,
                                                    float* __restrict__ Y1,
                                                    float* __restrict__ Y2t) {
    const int lane = threadIdx.x;
    const int n0 = blockIdx.x * 16;
    const int o0 = blockIdx.y * 16;
    const int b  = blockIdx.z;
    const int m  = lane & 15;            // A: M index / B,D: N index
    const int kh = (lane >> 4) << 1;     // K sub-offset per lane half (0 or 2)
    const int rbase = (lane >> 4) * 8;   // D rows 0..7 or 8..15

    const float* xb = x + (size_t)b * CC * NN;
    const float* wr = W + (size_t)(o0 + m) * TWOC;

    v8f acc1 = {};  // (W1+W2) * x
    v8f acc2 = {};  // W2 * x
#pragma unroll 4
    for (int k = 0; k < CC; k += 4) {
        const int ka = k + kh;
        const int kb = ka + 1;
        float w1a = wr[ka],      w1b = wr[kb];
        float w2a = wr[CC + ka], w2b = wr[CC + kb];
        v2f aS; aS.x = w1a + w2a; aS.y = w1b + w2b;
        v2f a2; a2.x = w2a;       a2.y = w2b;
        v2f bx; bx.x = xb[(size_t)ka * NN + n0 + m];
                bx.y = xb[(size_t)kb * NN + n0 + m];
        acc1 = __builtin_amdgcn_wmma_f32_16x16x4_f32(false, aS, false, bx,
                                                     (short)0, acc1, false, false);
        acc2 = __builtin_amdgcn_wmma_f32_16x16x4_f32(false, a2, false, bx,
                                                     (short)0, acc2, false, false);
    }
#pragma unroll
    for (int r = 0; r < 8; ++r) {
        int o = o0 + rbase + r;
        int n = n0 + m;
        Y1 [(size_t)b * OUTC * NN + (size_t)o * NN + n]  = acc1[r] + bias[o];
        Y2t[(size_t)b * NN * OUTC + (size_t)n * OUTC + o] = acc2[r];
    }
}

// ---------------------------------------------------------------------------
// Kernel 2: fused distance GEMM + per-row top-20 indices.
// rank score s[n,m] = 2*<x_n,x_m> - ||x_m||^2  (row-constant -||x_n||^2 dropped).
// Workgroup = 4 waves, handles 16 rows; waves stride over 512 column tiles.
// ---------------------------------------------------------------------------
__global__ void __launch_bounds__(128) knn_kernel(const float* __restrict__ x,
                                                  const float* __restrict__ xx,
                                                  int* __restrict__ idxOut) {
    const int tid  = threadIdx.x;
    const int wave = tid >> 5;
    const int lane = tid & 31;
    const int n0 = blockIdx.x * 16;
    const int b  = blockIdx.y;

    const float* xb  = x  + (size_t)b * CC * NN;
    const float* xxb = xx + (size_t)b * NN;

    __shared__ float s_score[4][16][17];       // per-wave 16x16 score tile (padded)
    __shared__ float s_cval[16][8][KNN];       // merge candidates per row
    __shared__ int   s_cidx[16][8][KNN];

    const int m  = lane & 15;
    const int kh = (lane >> 4) << 1;
    const int rbase = (lane >> 4) * 8;

    // Cache A tile (16 rows x 128 K) in registers: 32 x v2f = 64 VGPRs
    v2f a[32];
#pragma unroll
    for (int t = 0; t < 32; ++t) {
        int k = 4 * t + kh;
        a[t].x = xb[(size_t)k * NN + n0 + m];
        a[t].y = xb[(size_t)(k + 1) * NN + n0 + m];
    }

    // Per-lane sorted (descending) top-20 over this lane's column subset
    float tv[KNN]; int ti[KNN];
#pragma unroll
    for (int j = 0; j < KNN; ++j) { tv[j] = NEG_INF; ti[j] = 0; }

    const int r2 = lane >> 1;          // row scanned in selection phase
    const int ch = (lane & 1) * 8;     // column half

    for (int ct = wave; ct < NN / 16; ct += 4) {
        const int c0 = ct * 16;
        v8f acc = {};
#pragma unroll
        for (int t = 0; t < 32; ++t) {
            int k = 4 * t + kh;
            v2f bx;
            bx.x = xb[(size_t)k * NN + c0 + m];
            bx.y = xb[(size_t)(k + 1) * NN + c0 + m];
            acc = __builtin_amdgcn_wmma_f32_16x16x4_f32(false, a[t], false, bx,
                                                        (short)0, acc, false, false);
        }
        const float xv = xxb[c0 + m];
#pragma unroll
        for (int r = 0; r < 8; ++r)
            s_score[wave][rbase + r][m] = 2.0f * acc[r] - xv;
        // same-wave LDS RAW: DS ops are in-order per wave; compiler inserts dscnt wait
#pragma unroll
        for (int j = 0; j < 8; ++j) {
            float v  = s_score[wave][r2][ch + j];
            int   id = c0 + ch + j;
            if (v > tv[KNN - 1]) {
#pragma unroll
                for (int i2 = 0; i2 < KNN; ++i2) {
                    if (v > tv[i2]) {
                        float t0 = tv[i2]; int t1 = ti[i2];
                        tv[i2] = v; ti[i2] = id; v = t0; id = t1;
                    }
                }
            }
        }
    }

    // Dump 8 sorted candidate lists per row, then head-merge top 20
    const int slot = wave * 2 + (lane & 1);
#pragma unroll
    for (int j = 0; j < KNN; ++j) {
        s_cval[r2][slot][j] = tv[j];
        s_cidx[r2][slot][j] = ti[j];
    }
    __syncthreads();

    if (tid < 16) {
        int head[8];
#pragma unroll
        for (int s = 0; s < 8; ++s) head[s] = 0;
        int* op = idxOut + ((size_t)b * NN + (n0 + tid)) * KNN;
        for (int t = 0; t < KNN; ++t) {
            float best = NEG_INF; int bs = 0;
#pragma unroll
            for (int s = 0; s < 8; ++s) {
                float v = s_cval[tid][s][head[s]];
                if (v > best) { best = v; bs = s; }
            }
            op[t] = s_cidx[tid][bs][head[bs]];
#pragma unroll
            for (int s = 0; s < 8; ++s) head[s] += (s == bs) ? 1 : 0;
        }
    }
}

// ---------------------------------------------------------------------------
// Kernel 3: out[b,o,n] = relu(Y1[b,o,n] - min_j Y2t[b, idx[b,n,j], o])
// Threads vary n -> coalesced Y1 reads and out writes; gathers hit L2.
// ---------------------------------------------------------------------------
__global__ void __launch_bounds__(256) edge_out_kernel(const float* __restrict__ Y1,
                                                       const float* __restrict__ Y2t,
                                                       const int* __restrict__ idx,
                                                       float* __restrict__ out) {
    const int n = blockIdx.x * 256 + threadIdx.x;
    const int o = blockIdx.y;
    const int b = blockIdx.z;
    const int* ip = idx + ((size_t)b * NN + n) * KNN;
    const float* y2b = Y2t + (size_t)b * NN * OUTC + o;
    float mn = POS_INF;
#pragma unroll 4
    for (int j = 0; j < KNN; ++j) {
        int id = ip[j];
        mn = fminf(mn, y2b[(size_t)id * OUTC]);
    }
    const size_t off = (size_t)b * OUTC * NN + (size_t)o * NN + n;
    float h = Y1[off] - mn;
    out[off] = h > 0.f ? h : 0.f;
}

// ---------------------------------------------------------------------------
extern "C" void kernel_launch(void* const* d_in, const int* in_sizes, int n_in,
                              void* d_out, int out_size, void* d_ws, size_t ws_size,
                              hipStream_t stream) {
    (void)in_sizes; (void)n_in; (void)out_size; (void)ws_size;
    const float* x    = (const float*)d_in[0];   // [B,C,N]
    const float* W    = (const float*)d_in[1];   // [OUT,2C]
    const float* bias = (const float*)d_in[2];   // [OUT]
    float* out = (float*)d_out;                  // [B,OUT,N]

    // Workspace layout
    char* ws = (char*)d_ws;
    float* xx  = (float*)ws;                                  //  B*N         floats
    int*   idx = (int*)(ws + (size_t)BB * NN * 4);            //  B*N*K       ints
    float* Y1  = (float*)(ws + (size_t)BB * NN * 4
                             + (size_t)BB * NN * KNN * 4);    //  B*OUT*N     floats
    float* Y2t = Y1 + (size_t)BB * OUTC * NN;                 //  B*N*OUT     floats

    norms_kernel<<<dim3((BB * NN + 255) / 256), dim3(256), 0, stream>>>(x, xx);

    y_gemm_kernel<<<dim3(NN / 16, OUTC / 16, BB), dim3(32), 0, stream>>>(
        x, W, bias, Y1, Y2t);

    knn_kernel<<<dim3(NN / 16, BB), dim3(128), 0, stream>>>(x, xx, idx);

    edge_out_kernel<<<dim3(NN / 256, OUTC, BB), dim3(256), 0, stream>>>(
        Y1, Y2t, idx, out);
}